// RadialTensorProductLayer_48576080118162
// MI455X (gfx1250) — compile-verified
//
#include <hip/hip_runtime.h>

#define E_TOT 100000
#define NC 64
#define NRBF 32
#define NP 11
#define NW (NP * NC)        // 704 = 44 tiles of 16
#define TE 16               // edges per block
#define LDS_STRIDE 706      // 704 + 2: avoids bank overlap between half-waves

typedef float v2f __attribute__((ext_vector_type(2)));
typedef float v8f __attribute__((ext_vector_type(8)));
// 4-byte-aligned vector views so contiguous float groups lower to b96/b128
// VMEM ops (DWORD alignment mode handles align-4 multi-dword accesses).
typedef float v3f_a4 __attribute__((ext_vector_type(3), aligned(4)));
typedef float v4f_a4 __attribute__((ext_vector_type(4), aligned(4)));

__global__ __launch_bounds__(256)
void rtp_kernel(const float* __restrict__ x0, const float* __restrict__ x1,
                const float* __restrict__ x2, const float* __restrict__ y0,
                const float* __restrict__ y1, const float* __restrict__ y2,
                const float* __restrict__ rbf, const float* __restrict__ W,
                float* __restrict__ out0, float* __restrict__ out1,
                float* __restrict__ out2)
{
    __shared__ float w_lds[TE * LDS_STRIDE];

    const int tid  = threadIdx.x;
    const int lane = tid & 31;
    // Force wave index uniform -> scalar loop control, EXEC stays all-1s
    // around every WMMA (ISA requirement).
    const int wave = __builtin_amdgcn_readfirstlane(tid >> 5);
    const int e0   = blockIdx.x * TE;

    // ---------------- Phase 1: w = rbf @ W^T for 16 edges, via f32 WMMA ----
    {
        const int row   = lane & 15;          // A: M (edge row); B: N (out col)
        const int khalf = (lane >> 4) << 1;   // 0 for lanes 0-15, 2 for 16-31

        // A tile (16x32 rbf rows), per 16x16x4 f32 layout:
        //   VGPR0 = K=4t+khalf, VGPR1 = K=4t+khalf+1
        v2f a[8];
        const float* rrow = rbf + (size_t)(e0 + row) * NRBF;
        #pragma unroll
        for (int t = 0; t < 8; ++t)
            a[t] = *(const v2f*)(rrow + 4 * t + khalf);

        for (int nt = wave; nt < NW / 16; nt += 8) {
            const int n0 = nt * 16;
            const float* wrow = W + (size_t)(n0 + row) * NRBF;  // B[k,n] = W[n0+n, k]
            v8f acc = {0.f, 0.f, 0.f, 0.f, 0.f, 0.f, 0.f, 0.f};
            #pragma unroll
            for (int t = 0; t < 8; ++t) {
                v2f b = *(const v2f*)(wrow + 4 * t + khalf);
                acc = __builtin_amdgcn_wmma_f32_16x16x4_f32(
                        false, a[t], false, b, (short)0, acc, false, false);
            }
            // C/D layout: VGPR r, lanes 0-15 -> M=r, N=lane; lanes16-31 -> M=r+8
            const int mbase = (lane >> 4) << 3;
            #pragma unroll
            for (int r = 0; r < 8; ++r)
                w_lds[(mbase + r) * LDS_STRIDE + n0 + row] = acc[r];
        }
    }
    __syncthreads();

    // ---------------- Phase 2: fused tensor-product paths -------------------
    // Streaming data (touched exactly once) uses non-temporal loads/stores so
    // it does not evict the L2-resident W/rbf working set.
    #pragma unroll
    for (int i = 0; i < 4; ++i) {
        const int pair = i * 256 + tid;     // 0..1023
        const int el   = pair >> 6;         // local edge 0..15
        const int c    = pair & 63;         // channel (consecutive across lanes)
        const size_t ec = (size_t)(e0 + el) * NC + c;

        float w[NP];
        #pragma unroll
        for (int p = 0; p < NP; ++p)
            w[p] = w_lds[el * LDS_STRIDE + p * NC + c];

        const float x0s = __builtin_nontemporal_load(x0 + ec);
        const float y0s = __builtin_nontemporal_load(y0 + ec);

        v3f_a4 x1v = __builtin_nontemporal_load((const v3f_a4*)(x1 + ec * 3));
        v3f_a4 y1v = __builtin_nontemporal_load((const v3f_a4*)(y1 + ec * 3));

        float x2m[9], y2m[9];
        {
            v4f_a4 xa = __builtin_nontemporal_load((const v4f_a4*)(x2 + ec * 9));
            v4f_a4 xb = __builtin_nontemporal_load((const v4f_a4*)(x2 + ec * 9 + 4));
            float  xc = __builtin_nontemporal_load(x2 + ec * 9 + 8);
            v4f_a4 ya = __builtin_nontemporal_load((const v4f_a4*)(y2 + ec * 9));
            v4f_a4 yb = __builtin_nontemporal_load((const v4f_a4*)(y2 + ec * 9 + 4));
            float  yc = __builtin_nontemporal_load(y2 + ec * 9 + 8);
            #pragma unroll
            for (int q = 0; q < 4; ++q) { x2m[q] = xa[q]; x2m[4 + q] = xb[q]; }
            x2m[8] = xc;
            #pragma unroll
            for (int q = 0; q < 4; ++q) { y2m[q] = ya[q]; y2m[4 + q] = yb[q]; }
            y2m[8] = yc;
        }

        // zw = 0 : paths (0,0,0), (1,1,0), (2,2,0)
        float d11 = x1v[0]*y1v[0] + x1v[1]*y1v[1] + x1v[2]*y1v[2];
        float d22 = 0.f;
        #pragma unroll
        for (int q = 0; q < 9; ++q) d22 = fmaf(x2m[q], y2m[q], d22);
        float o0 = w[0] * (x0s * y0s) + w[4] * d11 + w[9] * d22;
        __builtin_nontemporal_store(o0, out0 + ec);

        // zw = 1 : paths (0,1,1), (1,0,1), (1,2,1), (2,1,1)
        v3f_a4 o1;
        #pragma unroll
        for (int q = 0; q < 3; ++q) {
            float p6 = x1v[0]*y2m[q] + x1v[1]*y2m[3 + q] + x1v[2]*y2m[6 + q]; // sum_i x1[i]*y2[i,q]
            float p8 = x2m[q*3+0]*y1v[0] + x2m[q*3+1]*y1v[1] + x2m[q*3+2]*y1v[2]; // sum_j x2[q,j]*y1[j]
            o1[q] = w[1] * (x0s * y1v[q]) + w[3] * (x1v[q] * y0s)
                  + w[6] * p6 + w[8] * p8;
        }
        __builtin_nontemporal_store(o1, (v3f_a4*)(out1 + ec * 3));

        // zw = 2 : paths (0,2,2), (1,1,2), (2,0,2), (2,2,2)
        float o2[9];
        #pragma unroll
        for (int r = 0; r < 3; ++r) {
            #pragma unroll
            for (int s = 0; s < 3; ++s) {
                float mm = x2m[r*3+0]*y2m[s] + x2m[r*3+1]*y2m[3 + s] + x2m[r*3+2]*y2m[6 + s];
                o2[r*3+s] = w[2]  * (x0s * y2m[r*3+s])
                          + w[5]  * (x1v[r] * y1v[s])
                          + w[7]  * (x2m[r*3+s] * y0s)
                          + w[10] * mm;
            }
        }
        {
            v4f_a4 oa, ob;
            #pragma unroll
            for (int q = 0; q < 4; ++q) { oa[q] = o2[q]; ob[q] = o2[4 + q]; }
            __builtin_nontemporal_store(oa, (v4f_a4*)(out2 + ec * 9));
            __builtin_nontemporal_store(ob, (v4f_a4*)(out2 + ec * 9 + 4));
            __builtin_nontemporal_store(o2[8], out2 + ec * 9 + 8);
        }
    }
}

extern "C" void kernel_launch(void* const* d_in, const int* in_sizes, int n_in,
                              void* d_out, int out_size, void* d_ws, size_t ws_size,
                              hipStream_t stream) {
    const float* x0  = (const float*)d_in[0];
    const float* x1  = (const float*)d_in[1];
    const float* x2  = (const float*)d_in[2];
    const float* y0  = (const float*)d_in[3];
    const float* y1  = (const float*)d_in[4];
    const float* y2  = (const float*)d_in[5];
    const float* rbf = (const float*)d_in[6];
    const float* W   = (const float*)d_in[7];

    float* out  = (float*)d_out;
    float* out0 = out;                                   // (E, NC)
    float* out1 = out + (size_t)E_TOT * NC;              // (E, NC, 3)
    float* out2 = out + (size_t)E_TOT * NC * 4;          // (E, NC, 3, 3)

    rtp_kernel<<<dim3(E_TOT / TE), dim3(256), 0, stream>>>(
        x0, x1, x2, y0, y1, y2, rbf, W, out0, out1, out2);
}